// RelativePositionalEncoding_32366873543059
// MI455X (gfx1250) — compile-verified
//
#include <hip/hip_runtime.h>
#include <stdint.h>

// RelativePositionalEncoding for MI455X (gfx1250).
// out[b,i,j,0:8] = table[clip(index[b,i]-index[b,j]+32, 0, 64)]  where
// table[k][d] = W[d][k] + bias[d].
// Pure HBM-store-bound (512 MB out, ~22us floor @ 23.3 TB/s) -> LDS gather +
// coalesced non-temporal b128 stores. W is staged into LDS via the Tensor
// Data Mover (tensor_load_to_lds + s_wait_tensorcnt).

#define MAX_REL_IDX 32
#define N_REL_POS   65       // 2*MAX_REL_IDX + 1
#define ATTN_DIM    8
#define W_ELEMS     (ATTN_DIM * N_REL_POS)   // 520 floats

typedef unsigned int u32x4 __attribute__((ext_vector_type(4)));
typedef int          i32x8 __attribute__((ext_vector_type(8)));
typedef int          i32x4 __attribute__((ext_vector_type(4)));
typedef float        f32x4 __attribute__((ext_vector_type(4)));

__global__ __launch_bounds__(256) void relpos_enc_kernel(
    const int*   __restrict__ index,   // [B, N]
    const float* __restrict__ W,       // [ATTN_DIM, N_REL_POS]
    const float* __restrict__ bias,    // [ATTN_DIM]
    float*       __restrict__ out,     // [B, N, N, ATTN_DIM]
    int N) {
  // Raw W staged by TDM, then biased/transposed gather table [k][d].
  __shared__ __align__(16) float ldsW[W_ELEMS];
  __shared__ __align__(16) float table[W_ELEMS];

  const int tid = threadIdx.x;
  const int blk = blockIdx.x;          // flat (b*N + i)
  const int b   = blk / N;

  // ---- Stage W (520 f32, contiguous) into LDS with the Tensor Data Mover.
  // Wave 0 only (wave-uniform branch; TDM is a wave-level op, EXEC ignored).
  if (tid < warpSize) {
    const uint64_t ga      = (uint64_t)(uintptr_t)W;
    const unsigned ldsAddr = (unsigned)(uintptr_t)(void*)ldsW; // low 32b = LDS byte offset

    // D# group 0 (CDNA5 ISA):
    //   [1:0] count=1 (valid user descriptor), [63:32] lds_addr,
    //   [120:64] global_addr, [127:126] type=2 ("image").
    u32x4 g0;
    g0.x = 1u;
    g0.y = ldsAddr;
    g0.z = (unsigned)(ga & 0xFFFFFFFFu);
    g0.w = (unsigned)((ga >> 32) & 0x1FFFFFFu) | (2u << 30);

    // D# group 1 (§8.4): workgroup_mask=0 (not in cluster),
    //   data_size=2 (4 bytes), tensor_dim0=520, tensor_dim1=1,
    //   tile_dim0=520, tile_dim1=1, tensor_dim0_stride=520.
    i32x8 g1;
    g1[0] = (int)(2u << 16);                    // data_size=4B; no pad/iterate/atomic
    g1[1] = (int)((unsigned)W_ELEMS << 16);     // tensor_dim0[15:0] at bits 63:48
    g1[2] = (int)(1u << 16);                    // tensor_dim0[31:16]=0; tensor_dim1=1
    g1[3] = (int)((unsigned)W_ELEMS << 16);     // tile_dim0 at bits 127:112
    g1[4] = 1;                                  // tile_dim1=1, tile_dim2=0
    g1[5] = W_ELEMS;                            // tensor_dim0_stride (data_size units)
    g1[6] = 0;
    g1[7] = 0;

    i32x4 z4 = {0, 0, 0, 0};                    // groups 2/3 unused (<=2D tensor)
    i32x8 z8 = {0, 0, 0, 0, 0, 0, 0, 0};        // unused trailing descriptor group
    __builtin_amdgcn_tensor_load_to_lds(g0, g1, z4, z4, z8, /*cpol=*/0);
    __builtin_amdgcn_s_wait_tensorcnt(0);       // TENSORcnt == 0 before LDS is read
  }
  __syncthreads();

  // ---- Build pre-biased, transposed table[k*8+d] = W[d][k] + bias[d].
  for (int t = tid; t < W_ELEMS; t += blockDim.x) {
    const int k = t >> 3;
    const int d = t & 7;
    table[t] = ldsW[d * N_REL_POS + k] + bias[d];
  }
  __syncthreads();

  // ---- Main stream: one (b,i) row per block; gather + NT b128 store pairs.
  const int  idx_i  = index[blk];                       // block-uniform (s_load)
  const int* idxRow = index + (size_t)b * N;
  float*     orow   = out + (size_t)blk * N * ATTN_DIM;

  for (int j = tid; j < N; j += blockDim.x) {
    int c = (idx_i - idxRow[j]) + MAX_REL_IDX;
    c = c < 0 ? 0 : (c > (N_REL_POS - 1) ? (N_REL_POS - 1) : c);

    const f32x4* row = (const f32x4*)(table + c * ATTN_DIM);  // 32B aligned
    const f32x4  lo  = row[0];
    const f32x4  hi  = row[1];

    f32x4* dst = (f32x4*)(orow + (size_t)j * ATTN_DIM);       // coalesced, 32B/lane
    __builtin_nontemporal_store(lo, dst);        // 512MB >> L2: keep it out of cache
    __builtin_nontemporal_store(hi, dst + 1);
  }
}

extern "C" void kernel_launch(void* const* d_in, const int* in_sizes, int n_in,
                              void* d_out, int out_size, void* d_ws, size_t ws_size,
                              hipStream_t stream) {
  (void)n_in; (void)d_ws; (void)ws_size;
  const int*   index = (const int*)d_in[0];    // [B, N] int32
  const float* W     = (const float*)d_in[1];  // [8, 65] f32
  const float* bias  = (const float*)d_in[2];  // [8] f32
  float*       out   = (float*)d_out;          // [B, N, N, 8] f32

  const long long BN = in_sizes[0];                                   // B*N
  const long long N  = (long long)out_size / (BN * (long long)ATTN_DIM);
  // One block per (b, i) row: 8192 blocks x 256 threads, 8 j's per thread.
  relpos_enc_kernel<<<dim3((unsigned)BN), dim3(256), 0, stream>>>(
      index, W, bias, out, (int)N);
}